// CritiGraph_83708912599576
// MI455X (gfx1250) — compile-verified
//
#include <hip/hip_runtime.h>
#include <hip/hip_bf16.h>

// ---------------- problem constants (from reference) ----------------
#define T1C   128
#define T2C   512
#define DC    256
#define TPC   8
#define HC    16
#define KC    4
#define CC    129   // 2*K*H + 1
#define OUTN  (T1C * CC * TPC)   // 132096

typedef float v2f __attribute__((ext_vector_type(2)));
typedef float v8f __attribute__((ext_vector_type(8)));

// ---------------------------------------------------------------
// Kernel 1: row-normalize emb_T1 (rows 0..127) and emb_T2 (rows 128..639)
// into a contiguous [640 x 256] f32 buffer. One block per row.
// ---------------------------------------------------------------
__global__ __launch_bounds__(256) void cg_normalize_kernel(
    const float* __restrict__ e1, const float* __restrict__ e2,
    float* __restrict__ out)
{
    __shared__ float red[256];
    const int row = blockIdx.x;              // 0..639
    const int t   = threadIdx.x;             // 0..255 == column
    const float* src = (row < T1C) ? (e1 + row * DC)
                                   : (e2 + (row - T1C) * DC);
    float x = src[t];
    red[t] = x * x;
    __syncthreads();
    for (int s = 128; s > 0; s >>= 1) {
        if (t < s) red[t] += red[t + s];
        __syncthreads();
    }
    float inv = 1.0f / sqrtf(red[0]);
    out[row * DC + t] = x * inv;
}

// ---------------------------------------------------------------
// Kernel 2: val_v = e1n @ e2n^T  (128x512, K=256) with f32 WMMA.
// One wave32 per 16x16 output tile; 8x32 = 256 tiles.
// A frag (16x4 f32): lane m=lane&15, half=lane>>4 -> {K=k0+2h, k0+2h+1}
// B frag (4x16 f32): B[k][n] = e2n[n][k] -> identical addressing on e2n rows.
// ---------------------------------------------------------------
__global__ __launch_bounds__(128) void cg_gemm_wmma_kernel(
    const float* __restrict__ A,   // e1n: 128x256
    const float* __restrict__ B,   // e2n: 512x256
    float* __restrict__ Cout)      // val_v: 128x512
{
    const int wave   = (blockIdx.x * blockDim.x + threadIdx.x) >> 5; // 0..255
    const int lane   = threadIdx.x & 31;
    const int tile_m = wave >> 5;          // 0..7
    const int tile_n = wave & 31;          // 0..31
    const int m      = lane & 15;
    const int half   = lane >> 4;

    const float* arow = A + (tile_m * 16 + m) * DC;
    const float* brow = B + (tile_n * 16 + m) * DC;

    v8f acc = {};
    for (int k0 = 0; k0 < DC; k0 += 4) {
        v2f a = *(const v2f*)(arow + k0 + 2 * half);
        v2f b = *(const v2f*)(brow + k0 + 2 * half);
        acc = __builtin_amdgcn_wmma_f32_16x16x4_f32(
            /*neg_a=*/false, a, /*neg_b=*/false, b,
            /*c_mod=*/(short)0, acc, /*reuse_a=*/false, /*reuse_b=*/false);
    }
    // C/D layout: VGPR r, lanes 0-15 -> M=r, lanes 16-31 -> M=r+8; N = m
#pragma unroll
    for (int r = 0; r < 8; ++r) {
        int M = tile_m * 16 + r + 8 * half;
        int N = tile_n * 16 + m;
        Cout[M * T2C + N] = acc[r];
    }
}

// ---------------------------------------------------------------
// Kernel 3: candidate generation + distances + masked MSE reduction
// over a T2 chunk. Grid = (T1, nsplit); block = 256 threads = 8 wave32.
// partial[chunk][t1][c][tp] = sum_{t2 in chunk} mask * (base + sg*mag)^2
//   base[t2][tp] = (dsum - ds[tp])/8 - val_v
//   mag = 0.125 - e/128,  e = 32 - clz(|c1|^|c2| + 1)   (== frexp exponent)
// ---------------------------------------------------------------
__global__ __launch_bounds__(256) void cg_loss_kernel(
    const int* __restrict__ sta_loc,       // (T1, TP)
    const int* __restrict__ pos_loc,       // (T2, TP)
    const unsigned char* __restrict__ mask,// (T1, T2) bool bytes
    const int* __restrict__ random_masks,  // (T1, H, K, TP)
    const float* __restrict__ valv,        // (T1, T2)
    float* __restrict__ dst,               // (nsplit, T1, C, TP) partials
    int t2_chunk)                          // T2C / nsplit
{
    __shared__ int   s_sta[TPC];
    __shared__ int   s_cnc[CC * TPC];
    __shared__ int   s_pos[T2C * TPC];
    __shared__ float s_base[T2C * TPC];
    __shared__ float s_mask[T2C];

    const int tid = threadIdx.x;
    const int t1  = blockIdx.x;
    const int t2b = blockIdx.y * t2_chunk;
    const int t2e = t2b + t2_chunk;

    if (tid < TPC) s_sta[tid] = sta_loc[t1 * TPC + tid];
    __syncthreads();

    // ---- candidate generation: c<64: res[j=c], c==64: sta, c>64: -res[j=c-65]
    for (int idx = tid; idx < CC * TPC; idx += 256) {
        int c  = idx >> 3;
        int tp = idx & 7;
        int sta = s_sta[tp];
        int v;
        if (c == 64) {
            v = sta;
        } else {
            int j = (c < 64) ? c : (c - 65);
            int i = j >> 2;          // H index
            int k = j & 3;           // K index
            int flip = 1 << i;
            int low  = random_masks[((t1 * HC + i) * KC + k) * TPC + tp] & (flip - 1);
            int res  = (sta ^ flip) ^ low;
            v = (c < 64) ? res : -res;
        }
        s_cnc[idx] = v;
    }

    // ---- dis_sta into s_base (temp), pos_loc into LDS (chunk range only)
    for (int idx = tid + t2b * TPC; idx < t2e * TPC; idx += 256) {
        int tp = idx & 7;
        int p  = pos_loc[idx];
        s_pos[idx] = p;
        int s1 = s_sta[tp];
        int a1 = abs(s1), a2 = abs(p);
        int xr = a1 ^ a2;
        int e  = 32 - __builtin_clz((unsigned)(xr + 1));
        float d = 1.0f - (float)e * (1.0f / 16.0f);
        bool same = (s1 >= 0) == (p >= 0);
        s_base[idx] = same ? d : -d;
    }
    __syncthreads();

    // ---- base[t2][tp] = (dsum - ds[tp])/8 - val_v[t1][t2] (in place)
    for (int t2 = t2b + tid; t2 < t2e; t2 += 256) {
        float ds[TPC];
        float dsum = 0.0f;
#pragma unroll
        for (int tp = 0; tp < TPC; ++tp) { ds[tp] = s_base[t2 * TPC + tp]; dsum += ds[tp]; }
        float vv = valv[t1 * T2C + t2];
#pragma unroll
        for (int tp = 0; tp < TPC; ++tp)
            s_base[t2 * TPC + tp] = (dsum - ds[tp]) * 0.125f - vv;
        s_mask[t2] = mask[t1 * T2C + t2] ? 1.0f : 0.0f;
    }
    __syncthreads();

    // ---- main accumulation: thread owns tp = tid&7, candidates c0+32r
    const int tp = tid & 7;
    const int c0 = tid >> 3;   // 0..31
    int   a1[5];
    bool  neg1[5];
    float acc[5];
    int nc = 0;
#pragma unroll
    for (int r = 0; r < 5; ++r) {
        int c = c0 + 32 * r;
        int v = (c < CC) ? s_cnc[c * TPC + tp] : 0;
        if (c < CC) nc = r + 1;
        a1[r]   = abs(v);
        neg1[r] = (v < 0);
        acc[r]  = 0.0f;
    }

    for (int t2 = t2b; t2 < t2e; ++t2) {
        float mk = s_mask[t2];
        if (mk == 0.0f) continue;       // uniform across block -> scalar branch
        int   p   = s_pos[t2 * TPC + tp];
        float bse = s_base[t2 * TPC + tp];
        int  a2   = abs(p);
        bool neg2 = (p < 0);
#pragma unroll
        for (int r = 0; r < 5; ++r) {
            if (r < nc) {
                int xr = a1[r] ^ a2;
                int e  = 32 - __builtin_clz((unsigned)(xr + 1));
                float mag  = 0.125f - (float)e * 0.0078125f;  // (1 - e/16)/8
                float diff = bse + ((neg1[r] == neg2) ? mag : -mag);
                acc[r] += diff * diff;
            }
        }
    }

    float* blk = dst + (size_t)blockIdx.y * OUTN;
#pragma unroll
    for (int r = 0; r < 5; ++r) {
        int c = c0 + 32 * r;
        if (c < CC) blk[(t1 * CC + c) * TPC + tp] = acc[r];
    }
}

// ---------------------------------------------------------------
// Kernel 4: reduce nsplit partial buffers into d_out.
// ---------------------------------------------------------------
__global__ __launch_bounds__(256) void cg_reduce_kernel(
    const float* __restrict__ partial, float* __restrict__ out, int nsplit)
{
    int i = blockIdx.x * 256 + threadIdx.x;
    if (i >= OUTN) return;
    float s = 0.0f;
    for (int c = 0; c < nsplit; ++c) s += partial[(size_t)c * OUTN + i];
    out[i] = s;
}

// ---------------------------------------------------------------
extern "C" void kernel_launch(void* const* d_in, const int* in_sizes, int n_in,
                              void* d_out, int out_size, void* d_ws, size_t ws_size,
                              hipStream_t stream) {
    const float* emb_T1       = (const float*)d_in[0];
    const float* emb_T2       = (const float*)d_in[1];
    const int*   sta_loc      = (const int*)d_in[2];
    const int*   pos_loc      = (const int*)d_in[3];
    const unsigned char* mask = (const unsigned char*)d_in[4];
    const int*   random_masks = (const int*)d_in[5];
    float*       out          = (float*)d_out;

    float* ws_f = (float*)d_ws;
    float* e1n     = ws_f;                        // 128*256
    float* e2n     = ws_f + T1C * DC;             // 512*256
    float* valv    = ws_f + (T1C + T2C) * DC;     // 128*512
    float* partial = valv + T1C * T2C;            // nsplit * OUTN

    // 1) normalize rows of both embedding matrices
    cg_normalize_kernel<<<T1C + T2C, 256, 0, stream>>>(emb_T1, emb_T2, e1n);

    // 2) val_v = e1n @ e2n^T via V_WMMA_F32_16X16X4_F32 (256 wave-tiles)
    cg_gemm_wmma_kernel<<<64, 128, 0, stream>>>(e1n, e2n, valv);

    // 3) loss with T2 split for occupancy (512 WGs / 4096 wave32s),
    //    falling back to direct single-pass if workspace is tight.
    const size_t base_floats = (size_t)(T1C + T2C) * DC + (size_t)T1C * T2C;
    const int    nsplit      = 4;
    const bool   can_split   =
        ws_size >= (base_floats + (size_t)nsplit * OUTN) * sizeof(float);

    if (can_split) {
        dim3 grid(T1C, nsplit);
        cg_loss_kernel<<<grid, 256, 0, stream>>>(sta_loc, pos_loc, mask,
                                                 random_masks, valv, partial,
                                                 T2C / nsplit);
        cg_reduce_kernel<<<(OUTN + 255) / 256, 256, 0, stream>>>(partial, out, nsplit);
    } else {
        dim3 grid(T1C, 1);
        cg_loss_kernel<<<grid, 256, 0, stream>>>(sta_loc, pos_loc, mask,
                                                 random_masks, valv, out,
                                                 T2C);
    }
}